// GCNConv_17841294148275
// MI455X (gfx1250) — compile-verified
//
#include <hip/hip_runtime.h>
#include <hip/hip_bf16.h>

typedef __attribute__((ext_vector_type(2))) float v2f;
typedef __attribute__((ext_vector_type(8))) float v8f;

#define NV     5000        // n_vertex (GEMM2 M and K)
#define HC     12288       // B*T*c_out = 16*12*64  (GEMM2 N)
#define MR     960000      // x_flat rows = B*C_IN*T*N/64

// gfx1250 async global->LDS staging (ASYNCcnt path), guarded so the file still
// compiles if this toolchain doesn't declare the builtins.
#if defined(__has_builtin)
#if __has_builtin(__builtin_amdgcn_global_load_async_to_lds_b128) && \
    __has_builtin(__builtin_amdgcn_s_wait_asynccnt)
#define USE_ASYNC_LDS 1
#endif
#endif

#ifdef USE_ASYNC_LDS
typedef int vi4_t __attribute__((vector_size(16)));
#define ASYNC_B128(gp, lp) \
    __builtin_amdgcn_global_load_async_to_lds_b128( \
        (vi4_t*)(void*)(gp), (vi4_t*)(void*)(lp), 0, 0)
#define ASYNC_WAIT() __builtin_amdgcn_s_wait_asynccnt(0)
#else
#define ASYNC_WAIT()
#endif

// ---------------------------------------------------------------------------
// Kernel 1: h[960000 x 64] = x_flat[960000 x 64] @ weight[64 x 64]
// 256 threads = 8 waves; each wave: 16 rows x 64 cols = 4 WMMA tiles, K=64.
// ---------------------------------------------------------------------------
__global__ __launch_bounds__(256) void gcn_k1_xw(const float* __restrict__ x,
                                                 const float* __restrict__ w,
                                                 float* __restrict__ h) {
    __shared__ float Xs[128][68];   // pad 4 -> conflict-free A-frag ds_load_b64
    __shared__ float Ws[64][72];    // pad 8 -> conflict-free B-frag gathers

    const int t    = threadIdx.x;
    const int lane = t & 31;
    const int wv   = t >> 5;                       // wave 0..7
    const long base = (long)blockIdx.x * (128 * 64);

    // weight 64x64 -> LDS (coalesced scalars)
    #pragma unroll
    for (int i = 0; i < 16; ++i) {
        int f = t + i * 256;
        Ws[f >> 6][f & 63] = w[f];
    }
    // x tile 128x64 (contiguous 8192 floats) -> LDS, 16B granules
#ifdef USE_ASYNC_LDS
    #pragma unroll
    for (int i = 0; i < 8; ++i) {
        int f4 = t + i * 256;                      // 0..2047
        int r = f4 >> 4;                           // 16 float4 per row
        int c = (f4 & 15) << 2;
        ASYNC_B128(x + base + (long)f4 * 4, &Xs[r][c]);
    }
    ASYNC_WAIT();
#else
    const float4* x4 = (const float4*)(x + base);
    #pragma unroll
    for (int i = 0; i < 8; ++i) {
        int f4 = t + i * 256;
        float4 v = x4[f4];
        int r = f4 >> 4;
        int c = (f4 & 15) << 2;
        *(float4*)&Xs[r][c] = v;
    }
#endif
    __syncthreads();

    v8f acc[4] = {};
    const int mlane = lane & 15;
    const int khalf = (lane >> 4) << 1;            // 0 (lanes 0-15) or 2
    const int m     = (wv << 4) + mlane;

    #pragma unroll
    for (int k0 = 0; k0 < 64; k0 += 4) {
        // A frag: VGPR0={K0|K2}, VGPR1={K1|K3}; row stride 68 -> 8B aligned
        v2f a = *(const v2f*)&Xs[m][k0 + khalf];
        #pragma unroll
        for (int nt = 0; nt < 4; ++nt) {
            int kr = k0 + khalf;
            int n  = (nt << 4) + mlane;
            v2f b;
            b.x = Ws[kr][n];
            b.y = Ws[kr + 1][n];
            acc[nt] = __builtin_amdgcn_wmma_f32_16x16x4_f32(
                false, a, false, b, (short)0, acc[nt], false, false);
        }
    }

    const int rhalf = (lane >> 4) << 3;            // 0 or 8 (C/D layout)
    const long rb   = (long)blockIdx.x * 128 + (wv << 4) + rhalf;
    #pragma unroll
    for (int nt = 0; nt < 4; ++nt) {
        int n = (nt << 4) + mlane;
        #pragma unroll
        for (int j = 0; j < 8; ++j) {
            h[(rb + j) * 64 + n] = acc[nt][j];
        }
    }
}

// ---------------------------------------------------------------------------
// Kernel 2: out[5000 x 12288] = filter[5000 x 5000] @ h[5000 x 12288] + bias
// BM=BN=128, BK=32, double-buffered LDS with async DMA prefetch of the next
// k-block overlapped with the 64-WMMA compute phase. One barrier/iteration.
// ---------------------------------------------------------------------------
__global__ __launch_bounds__(256) void gcn_k2_spmm(const float* __restrict__ A,
                                                   const float* __restrict__ Hm,
                                                   const float* __restrict__ bias,
                                                   float* __restrict__ out) {
    __shared__ float As[2][128][36];  // stride 36: 16B-aligned rows, conflict-free
    __shared__ float Bs[2][32][144];  // stride 144: 16B-aligned, half-waves disjoint

    const int t    = threadIdx.x;
    const int lane = t & 31;
    const int wv   = t >> 5;
    const int wr   = wv & 3;                  // m offset = wr*32 (2 tiles)
    const int wc   = wv >> 2;                 // n offset = wc*64 (4 tiles)
    const int blockM = blockIdx.y * 128;
    const int blockN = blockIdx.x * 128;

    const int arow0 = t >> 3;                 // As: rows arow0 + i*32
    const int acol  = (t & 7) << 2;           //     cols acol..acol+3
    const int brow0 = t >> 5;                 // Bs: rows brow0 + i*8
    const int bcol  = (t & 31) << 2;          //     cols bcol..bcol+3

    const int mlane = lane & 15;
    const int khalf = (lane >> 4) << 1;       // 0 or 2

    v8f acc[2][4] = {};

    // Stage one 32-deep k-block into LDS buffer `buf`.
    auto load_block = [&](int buf, int kb) {
        if (kb + 32 <= NV) {                  // full block
#ifdef USE_ASYNC_LDS
            #pragma unroll
            for (int i = 0; i < 4; ++i) {
                int r  = arow0 + i * 32;
                int gr = blockM + r; if (gr > NV - 1) gr = NV - 1;   // clamp M tail
                ASYNC_B128(&A[(long)gr * NV + kb + acol], &As[buf][r][acol]);
            }
            #pragma unroll
            for (int i = 0; i < 4; ++i) {
                int r = brow0 + i * 8;
                ASYNC_B128(&Hm[(long)(kb + r) * HC + blockN + bcol], &Bs[buf][r][bcol]);
            }
#else
            #pragma unroll
            for (int i = 0; i < 4; ++i) {
                int r  = arow0 + i * 32;
                int gr = blockM + r; if (gr > NV - 1) gr = NV - 1;
                float4 v = *(const float4*)&A[(long)gr * NV + kb + acol];
                *(float4*)&As[buf][r][acol] = v;
            }
            #pragma unroll
            for (int i = 0; i < 4; ++i) {
                int r = brow0 + i * 8;
                float4 v = *(const float4*)&Hm[(long)(kb + r) * HC + blockN + bcol];
                *(float4*)&Bs[buf][r][bcol] = v;
            }
#endif
        } else {                              // K tail: zero-fill out-of-range
            #pragma unroll
            for (int i = 0; i < 4; ++i) {
                int r  = arow0 + i * 32;
                int gr = blockM + r; if (gr > NV - 1) gr = NV - 1;
                #pragma unroll
                for (int j = 0; j < 4; ++j) {
                    int k = kb + acol + j;
                    As[buf][r][acol + j] = (k < NV) ? A[(long)gr * NV + k] : 0.0f;
                }
            }
            #pragma unroll
            for (int i = 0; i < 4; ++i) {
                int r = brow0 + i * 8;
                int k = kb + r;
                #pragma unroll
                for (int j = 0; j < 4; ++j) {
                    Bs[buf][r][bcol + j] = (k < NV) ? Hm[(long)k * HC + blockN + bcol + j]
                                                    : 0.0f;
                }
            }
        }
    };

    const int nblocks = (NV + 31) / 32;       // 157 (last block is the 8-row tail)

    load_block(0, 0);                          // prologue

    for (int ib = 0; ib < nblocks; ++ib) {
        const int cur = ib & 1;

        ASYNC_WAIT();                          // loads for buf[cur] complete
        __syncthreads();                       // visible to all waves; buf[cur^1]
                                               // no longer read by anyone

        if (ib + 1 < nblocks) {
            load_block(cur ^ 1, (ib + 1) * 32);    // DMA overlaps compute below
            // warm L2 one more block ahead for the streamed h matrix
            if (ib + 2 < nblocks)
                __builtin_prefetch(&Hm[(long)((ib + 2) * 32 + brow0) * HC + blockN + bcol], 0, 0);
        }

        #pragma unroll
        for (int ks = 0; ks < 8; ++ks) {
            const int k0 = ks * 4;
            v2f a[2];
            #pragma unroll
            for (int mt = 0; mt < 2; ++mt)
                a[mt] = *(const v2f*)&As[cur][wr * 32 + mt * 16 + mlane][k0 + khalf];
            #pragma unroll
            for (int nt = 0; nt < 4; ++nt) {
                int kr = k0 + khalf;
                int n  = wc * 64 + nt * 16 + mlane;
                v2f b;
                b.x = Bs[cur][kr][n];
                b.y = Bs[cur][kr + 1][n];
                #pragma unroll
                for (int mt = 0; mt < 2; ++mt)
                    acc[mt][nt] = __builtin_amdgcn_wmma_f32_16x16x4_f32(
                        false, a[mt], false, b, (short)0, acc[mt][nt], false, false);
            }
        }
    }

    // epilogue: add bias (index = n % 64 since HC % 64 == 0), predicated M tail
    const int rhalf = (lane >> 4) << 3;
    #pragma unroll
    for (int mt = 0; mt < 2; ++mt) {
        #pragma unroll
        for (int nt = 0; nt < 4; ++nt) {
            int n   = blockN + wc * 64 + nt * 16 + mlane;
            float b = bias[n & 63];
            #pragma unroll
            for (int j = 0; j < 8; ++j) {
                int m = blockM + wr * 32 + mt * 16 + rhalf + j;
                if (m < NV)
                    out[(long)m * HC + n] = acc[mt][nt][j] + b;
            }
        }
    }
}

// ---------------------------------------------------------------------------
extern "C" void kernel_launch(void* const* d_in, const int* in_sizes, int n_in,
                              void* d_out, int out_size, void* d_ws, size_t ws_size,
                              hipStream_t stream) {
    (void)in_sizes; (void)n_in; (void)out_size; (void)ws_size;
    const float* x    = (const float*)d_in[0];  // [16,64,12,5000] raw-flat
    const float* filt = (const float*)d_in[1];  // [5000,5000]
    const float* w    = (const float*)d_in[2];  // [64,64]
    const float* bias = (const float*)d_in[3];  // [64]
    float* out = (float*)d_out;                 // [5000*12288]
    float* h   = (float*)d_ws;                  // needs 960000*64*4 = 245.76 MB

    gcn_k1_xw<<<MR / 128, 256, 0, stream>>>(x, w, h);
    dim3 g2(HC / 128, (NV + 127) / 128);        // 96 x 40
    gcn_k2_spmm<<<g2, 256, 0, stream>>>(filt, h, bias, out);
}